// BP_Decoder_30889404792871
// MI455X (gfx1250) — compile-verified
//
#include <hip/hip_runtime.h>
#include <stdint.h>

// LDPC (3,6)-regular BP decoder, one workgroup per batch row.
// v2: all index arrays compressed to u16 and cached in LDS once per block;
// reference permutations inverted into u16 scatter tables so each BP phase
// writes results directly in edge order (no permute passes).
// Global traffic per block = 1 row in + 1 row out, via CDNA5 async LDS DMA.

#define N_VN   2040
#define M_CN   1020
#define D_CN   6
#define D_VN   3
#define NEDGE  6120
#define CLIPV  1e-7f
#define NT     256
#define ROW_CHUNKS ((N_VN * 4) / 16)   // 510 x b128 per row (8160 B, 16B aligned)
#define INVALID16 0xFFFFu

// ---- CDNA5 async global<->LDS DMA (inline asm; SADDR form) -----------------

__device__ __forceinline__ unsigned lds_off_u32(const void* p) {
  // Flat LDS addresses carry the wave-relative LDS byte offset in addr[31:0].
  return (unsigned)(uintptr_t)p;
}

__device__ __forceinline__ void async_g2l_b128(unsigned lds_addr, unsigned voff,
                                               unsigned long long gbase) {
  asm volatile("global_load_async_to_lds_b128 %0, %1, %2"
               :: "v"(lds_addr), "v"(voff), "s"(gbase) : "memory");
}

__device__ __forceinline__ void async_l2g_b128(unsigned voff, unsigned lds_addr,
                                               unsigned long long gbase) {
  asm volatile("global_store_async_from_lds_b128 %0, %1, %2"
               :: "v"(voff), "v"(lds_addr), "s"(gbase) : "memory");
}

__device__ __forceinline__ void wait_asynccnt0() {
  asm volatile("s_wait_asynccnt 0" ::: "memory");
}

// ---------------------------------------------------------------------------

__global__ __launch_bounds__(NT)
void bp_decode_kernel(const float* __restrict__ llr_in,      // [B, N_VN] raw llr_demapper
                      const int*   __restrict__ cn_msg,      // [M_CN*D_CN]
                      const int*   __restrict__ vn_msg,      // [N_VN*D_VN]
                      const int*   __restrict__ vn2cn,       // [E]
                      const int*   __restrict__ cn_mask,     // [E]
                      const int*   __restrict__ vn_mask,     // [E]
                      const int*   __restrict__ edge_vn,     // [E]
                      const int*   __restrict__ nb_iter_p,   // [1]
                      float*       __restrict__ out)         // [B, N_VN]
{
  __shared__ __align__(16) float s_L[N_VN];       //  8160 B  (+llr_demapper; llr = -s_L)
  __shared__ __align__(16) float s_vc[NEDGE];     // 24480 B  var->check msgs, edge order
  __shared__ __align__(16) float s_cv[NEDGE];     // 24480 B  check->var msgs, edge order
  __shared__ __align__(16) unsigned short s_cnmsg[NEDGE];  // 12240 B  cn gather indices
  __shared__ __align__(16) unsigned short s_vnmsg[NEDGE];  // 12240 B  vn gather indices
  __shared__ __align__(16) unsigned short s_invcn[NEDGE];  // 12240 B  inv(cn_mask)
  __shared__ __align__(16) unsigned short s_invvn[NEDGE];  // 12240 B  inv(vn_mask . vn2cn)

  const int tid = threadIdx.x;
  const int b   = blockIdx.x;

  // --- DMA the LLR row into LDS (no VGPR staging) ---
  {
    const unsigned long long in_row =
        (unsigned long long)(uintptr_t)(llr_in + (size_t)b * N_VN);
    const unsigned l_base = lds_off_u32(s_L);
    for (int i = tid; i < ROW_CHUNKS; i += NT) {
      unsigned voff = (unsigned)i * 16u;
      async_g2l_b128(l_base + voff, voff, in_row);
    }
  }

  // --- cache index arrays in LDS as u16; init inverse tables (overlaps DMA) ---
  for (int f = tid; f < NEDGE; f += NT) {
    s_cnmsg[f] = (unsigned short)cn_msg[f];
    s_vnmsg[f] = (unsigned short)vn_msg[f];
    s_invcn[f] = (unsigned short)INVALID16;
    s_invvn[f] = (unsigned short)INVALID16;
  }
  wait_asynccnt0();      // each wave: its async chunks landed in LDS
  __syncthreads();

  // --- build inverse scatter tables + initial vc[e] = llr[edge_vn[e]] ---
  for (int e = tid; e < NEDGE; e += NT) {
    s_invcn[cn_mask[e]]         = (unsigned short)e;  // cv_edge[e] = cv_flat[cn_mask[e]]
    s_invvn[vn_mask[vn2cn[e]]]  = (unsigned short)e;  // vc_edge[e] = vc_flat[vn_mask[vn2cn[e]]]
    s_vc[e] = -s_L[edge_vn[e]];                       // llr = -L
  }
  __syncthreads();

  const int iters = nb_iter_p[0];

  // phase 0: initial check_to_var; phases 1..iters: var_to_check + check_to_var
  for (int phase = 0; phase <= iters; ++phase) {
    if (phase > 0) {
      // ---- var_to_check: vc_flat[vn*3+k] = (sum_m + llr) - m_k, scatter to edge order
      for (int vn = tid; vn < N_VN; vn += NT) {
        const int base = vn * D_VN;
        const unsigned i0 = s_vnmsg[base + 0];
        const unsigned i1 = s_vnmsg[base + 1];
        const unsigned i2 = s_vnmsg[base + 2];
        const float m0 = (i0 < NEDGE) ? s_cv[i0] : 0.0f;   // pad -> 0.0
        const float m1 = (i1 < NEDGE) ? s_cv[i1] : 0.0f;
        const float m2 = (i2 < NEDGE) ? s_cv[i2] : 0.0f;
        const float a = ((m0 + m1) + m2) - s_L[vn];        // sum + llr
        const unsigned p0 = s_invvn[base + 0];
        const unsigned p1 = s_invvn[base + 1];
        const unsigned p2 = s_invvn[base + 2];
        if (p0 != INVALID16) s_vc[p0] = a - m0;
        if (p1 != INVALID16) s_vc[p1] = a - m1;
        if (p2 != INVALID16) s_vc[p2] = a - m2;
      }
      __syncthreads();
    }

    // ---- check_to_var, scatter to edge order ----
    for (int cn = tid; cn < M_CN; cn += NT) {
      const int base = cn * D_CN;
      float t[D_CN], ex[D_CN];
      #pragma unroll
      for (int j = 0; j < D_CN; ++j) {
        const unsigned idx = s_cnmsg[base + j];
        t[j] = (idx < NEDGE) ? tanhf(0.5f * s_vc[idx]) : 1.0f;  // pad -> 1.0 (tanh domain)
      }
      // exclude-self product via prefix/suffix cumprod (reference order)
      float p = 1.0f;
      #pragma unroll
      for (int j = 0; j < D_CN; ++j) { ex[j] = p; p *= t[j]; }
      float s = 1.0f;
      #pragma unroll
      for (int j = D_CN - 1; j >= 0; --j) { ex[j] *= s; s *= t[j]; }
      #pragma unroll
      for (int j = 0; j < D_CN; ++j) {
        float y = ex[j];
        const float c = (y > 0.0f) ? CLIPV : ((y < 0.0f) ? -CLIPV : 0.0f);
        y -= c;                                    // y = excl - sign(excl)*CLIP
        const unsigned pos = s_invcn[base + j];
        // 2*atanh(y) = log1p(2y / (1-y))
        if (pos != INVALID16) s_cv[pos] = log1pf(2.0f * y / (1.0f - y));
      }
    }
    __syncthreads();
  }

  // ---- final: out[n] = -llr_dec = L[n] - sum_k cv ----  (computed into s_vc)
  for (int n = tid; n < N_VN; n += NT) {
    const int base = n * D_VN;
    const unsigned i0 = s_vnmsg[base + 0];
    const unsigned i1 = s_vnmsg[base + 1];
    const unsigned i2 = s_vnmsg[base + 2];
    const float m0 = (i0 < NEDGE) ? s_cv[i0] : 0.0f;
    const float m1 = (i1 < NEDGE) ? s_cv[i1] : 0.0f;
    const float m2 = (i2 < NEDGE) ? s_cv[i2] : 0.0f;
    s_vc[n] = s_L[n] - ((m0 + m1) + m2);
  }
  __syncthreads();

  // --- DMA result row LDS -> global ---
  {
    const unsigned long long out_row =
        (unsigned long long)(uintptr_t)(out + (size_t)b * N_VN);
    const unsigned v_base = lds_off_u32(s_vc);
    for (int i = tid; i < ROW_CHUNKS; i += NT) {
      unsigned voff = (unsigned)i * 16u;
      async_l2g_b128(voff, v_base + voff, out_row);
    }
    wait_asynccnt0();
  }
}

extern "C" void kernel_launch(void* const* d_in, const int* in_sizes, int n_in,
                              void* d_out, int out_size, void* d_ws, size_t ws_size,
                              hipStream_t stream) {
  const int*   cn_msg  = (const int*)d_in[0];
  const int*   vn_msg  = (const int*)d_in[1];
  const int*   vn2cn   = (const int*)d_in[2];
  const int*   cn_mask = (const int*)d_in[3];
  const int*   vn_mask = (const int*)d_in[4];
  const int*   edge_vn = (const int*)d_in[5];
  const float* llr     = (const float*)d_in[6];
  const int*   nb_iter = (const int*)d_in[7];
  float*       out     = (float*)d_out;

  const int B = in_sizes[6] / N_VN;   // 1024

  bp_decode_kernel<<<B, NT, 0, stream>>>(llr, cn_msg, vn_msg, vn2cn,
                                         cn_mask, vn_mask, edge_vn,
                                         nb_iter, out);
}